// Gene_Ontology_GNN_29033978921575
// MI455X (gfx1250) — compile-verified
//
#include <hip/hip_runtime.h>

#define DIM 256
#define LDS_STRIDE 260   // pad 16x256 A tile rows by 4 floats -> conflict-free ds_load_b64

typedef __attribute__((ext_vector_type(2))) float v2f;
typedef __attribute__((ext_vector_type(8))) float v8f;

// ---------------- degree / normalization ----------------

__global__ void zero_f32(float* __restrict__ p, int n) {
    int i = blockIdx.x * blockDim.x + threadIdx.x;
    if (i < n) p[i] = 0.f;
}

__global__ void deg_accum(const int* __restrict__ col, const float* __restrict__ w,
                          float* __restrict__ deg, int E) {
    int i = blockIdx.x * blockDim.x + threadIdx.x;
    if (i < E) atomicAdd(&deg[col[i]], w[i]);
}

__global__ void deg_to_dis(float* __restrict__ deg, int n) {
    int i = blockIdx.x * blockDim.x + threadIdx.x;
    if (i < n) {
        float d = deg[i];
        deg[i] = (d > 0.f) ? rsqrtf(d) : 0.f;   // deg_inv_sqrt, inf -> 0 (in place)
    }
}

// ---------------- dense GEMM: Y = emb @ lin_w^T  (fp32 WMMA) ----------------
// One block = one 16-row M tile. 16 waves, wave w owns N columns [16w, 16w+16).
// A tile (16x256) staged in LDS with row padding; B fragments read from lin_w
// (B[k][n] = lin_w[n][k], so a lane reads 2 consecutive k's of one lin_w row -> b64).

__global__ __launch_bounds__(512) void gemm_wmma(const float* __restrict__ A,
                                                 const float* __restrict__ W,
                                                 float* __restrict__ Y, int N) {
    __shared__ float As[16 * LDS_STRIDE];
    const int tid = threadIdx.x;
    const int m0  = blockIdx.x * 16;
    const bool full = (m0 + 16 <= N);   // wave-uniform: true for every block when N%16==0

    // cooperative, fully coalesced load of the 16x256 A tile (8 floats/thread)
    {
        int idx  = tid * 8;              // 0..4095
        int row  = idx >> 8;             // 0..15
        int ccol = idx & 255;
        float4 p0 = {0.f,0.f,0.f,0.f}, p1 = p0;
        if (full || (m0 + row < N)) {
            const float4* src = (const float4*)(A + (size_t)(m0 + row) * DIM + ccol);
            p0 = src[0];
            p1 = src[1];
        }
        float* dst = &As[row * LDS_STRIDE + ccol];
        *(float4*)(dst)     = p0;
        *(float4*)(dst + 4) = p1;
    }
    __syncthreads();

    const int wave = tid >> 5;
    const int lane = tid & 31;
    const int half = lane >> 4;     // 0: lanes 0-15 (K+0,K+1), 1: lanes 16-31 (K+2,K+3)
    const int r    = lane & 15;
    const int n0   = wave * 16;

    v8f acc = {};   // 16x16 fp32 accumulator (8 VGPRs)

    const float* arow = &As[r * LDS_STRIDE];
    const float* wrow = W + (size_t)(n0 + r) * DIM;   // lin_w[n0+r][*]

#pragma unroll 4
    for (int k = 0; k < DIM; k += 4) {
        int kk = k + 2 * half;
        v2f a = *(const v2f*)(arow + kk);   // A[r][kk], A[r][kk+1]      (ds_load_b64)
        v2f b = *(const v2f*)(wrow + kk);   // B[kk][n0+r], B[kk+1][..]  (global b64)
        acc = __builtin_amdgcn_wmma_f32_16x16x4_f32(
            /*neg_a=*/false, a, /*neg_b=*/false, b,
            /*c_mod=*/(short)0, acc, /*reuse_a=*/false, /*reuse_b=*/false);
    }

    // C/D layout: VGPR v -> row (v + 8*half), col = lane&15 within the tile
    float* yo = Y + (size_t)m0 * DIM + n0 + r;
    if (full) {
#pragma unroll
        for (int v = 0; v < 8; ++v)
            yo[(size_t)(v + 8 * half) * DIM] = acc[v];
    } else {
#pragma unroll
        for (int v = 0; v < 8; ++v) {
            int mrow = v + 8 * half;
            if (m0 + mrow < N) yo[(size_t)mrow * DIM] = acc[v];
        }
    }
}

// ---------------- output init: out[n][:] = lin_b ----------------

__global__ void init_bias(const float4* __restrict__ b4, float4* __restrict__ out4, int n4) {
    int i = blockIdx.x * blockDim.x + threadIdx.x;
    if (i < n4) out4[i] = b4[i & (DIM / 4 - 1)];
}

// ---------------- sparse scatter: out[col] += norm * Y[row] ----------------
// One wave per edge (grid-stride): lanes sweep the 256-wide feature row with
// two coalesced float4 (b128) loads; accumulation via f32 global atomics
// (resolve in L2; Y and out both fit in the 192MB L2).

__global__ void sgc_scatter(const int* __restrict__ ei, const float* __restrict__ w,
                            const float* __restrict__ dis, const float* __restrict__ Y,
                            float* __restrict__ out, int E) {
    const int lane = threadIdx.x & 31;
    int gw = (blockIdx.x * blockDim.x + threadIdx.x) >> 5;
    int nw = (gridDim.x * blockDim.x) >> 5;
    for (int e = gw; e < E; e += nw) {
        int r = ei[e];
        int c = ei[E + e];
        float norm = dis[r] * w[e] * dis[c];
        if (norm == 0.f) continue;
        const float4* ys = (const float4*)(Y + (size_t)r * DIM) + lane;   // 16B/lane, 512B/wave
        float*        od = out + (size_t)c * DIM + lane * 4;
#pragma unroll
        for (int ch = 0; ch < 2; ++ch) {                 // 2 x 128 floats
            float4 y = ys[ch * 32];
            float* o = od + ch * 128;
            atomicAdd(o + 0, norm * y.x);
            atomicAdd(o + 1, norm * y.y);
            atomicAdd(o + 2, norm * y.z);
            atomicAdd(o + 3, norm * y.w);
        }
    }
}

// ---------------- launch ----------------

extern "C" void kernel_launch(void* const* d_in, const int* in_sizes, int n_in,
                              void* d_out, int out_size, void* d_ws, size_t ws_size,
                              hipStream_t stream) {
    // inputs: 0:x (unused) 1:edge_index(2,E) 2:edge_weight(E) 3:emb(N,256) 4:lin_w(256,256) 5:lin_b(256)
    const int*   ei    = (const int*)d_in[1];
    const float* w     = (const float*)d_in[2];
    const float* emb   = (const float*)d_in[3];
    const float* lin_w = (const float*)d_in[4];
    const float* lin_b = (const float*)d_in[5];
    const int E = in_sizes[2];
    const int N = in_sizes[3] / DIM;

    float* Y   = (float*)d_ws;                                            // N*256 floats (51.2 MB)
    float* deg = (float*)((char*)d_ws + (size_t)N * DIM * sizeof(float)); // N floats; becomes dis
    float* out = (float*)d_out;

    zero_f32  <<<(N + 255) / 256, 256, 0, stream>>>(deg, N);
    deg_accum <<<(E + 255) / 256, 256, 0, stream>>>(ei + E, w, deg, E);
    deg_to_dis<<<(N + 255) / 256, 256, 0, stream>>>(deg, N);

    gemm_wmma<<<(N + 15) / 16, 512, 0, stream>>>(emb, lin_w, Y, N);

    const int n4 = N * (DIM / 4);
    init_bias<<<(n4 + 255) / 256, 256, 0, stream>>>((const float4*)lin_b, (float4*)out, n4);

    sgc_scatter<<<8192, 256, 0, stream>>>(ei, w, deg, Y, out, E);
}